// RecurrentNeuralNetwork_77902116815028
// MI455X (gfx1250) — compile-verified
//
#include <hip/hip_runtime.h>
#include <hip/hip_bf16.h>

// Problem constants (match the reference).
constexpr int B  = 128;
constexpr int T  = 2048;
constexpr int I  = 32;
constexpr int H  = 64;
constexpr int H2 = 2 * H;          // 128
constexpr int NROWS = B * T;       // 262144

typedef float v2f __attribute__((ext_vector_type(2)));
typedef float v8f __attribute__((ext_vector_type(8)));

// D = A(16x4 f32) * B(4x16 f32) + C(16x16 f32), exact fp32 WMMA.
static __device__ __forceinline__ v8f wmma4(v2f a, v2f b, v8f c) {
    return __builtin_amdgcn_wmma_f32_16x16x4_f32(
        /*neg_a=*/false, a, /*neg_b=*/false, b,
        /*c_mod=*/(short)0, c, /*reuse_a=*/false, /*reuse_b=*/false);
}

// ---------------------------------------------------------------------------
// Kernel 1: input projection. xw[t][b][dir*64 + h] = x[b,t,:] . W_ih[h,:] + bias
// One wave = 16 consecutive timesteps of one batch row, one direction.
// ---------------------------------------------------------------------------
__global__ void k_inproj(const float* __restrict__ x,
                         const float* __restrict__ Wf, const float* __restrict__ Wb,
                         const float* __restrict__ bif, const float* __restrict__ bhf,
                         const float* __restrict__ bib, const float* __restrict__ bhb,
                         float* __restrict__ ws) {
    const int lane  = threadIdx.x & 31;
    const int wid   = blockIdx.x * (blockDim.x >> 5) + (threadIdx.x >> 5);
    const int dir   = wid & 1;
    const int tile  = wid >> 1;              // 0..16383
    const int b     = tile >> 7;             // batch row
    const int t0    = (tile & 127) << 4;     // 16-aligned timestep base
    const int mrow  = lane & 15;
    const int hi    = lane >> 4;
    const int khalf = hi << 1;               // 0 or 2

    const float* W   = dir ? Wb  : Wf;
    const float* bi  = dir ? bib : bif;
    const float* bh  = dir ? bhb : bhf;

    // A fragments: x[b, t0+mrow, :]  (16 rows x K=32 -> 8 K-steps)
    v2f a[8];
    const float* xrow = x + ((size_t)b * T + (t0 + mrow)) * I;
#pragma unroll
    for (int ks = 0; ks < 8; ++ks)
        a[ks] = *(const v2f*)(xrow + ks * 4 + khalf);

#pragma unroll
    for (int nt = 0; nt < 4; ++nt) {
        const int col = nt * 16 + mrow;          // h index within direction
        const float bias = bi[col] + bh[col];
        v8f c;
#pragma unroll
        for (int v = 0; v < 8; ++v) c[v] = bias;
#pragma unroll
        for (int ks = 0; ks < 8; ++ks) {
            v2f bw = *(const v2f*)(W + (size_t)col * I + ks * 4 + khalf); // B[k][n]=W[n][k]
            c = wmma4(a[ks], bw, c);
        }
        // Store C: element (v + 8*hi, col) -> ws[(t0+r)*B + b][dir*H + col]
#pragma unroll
        for (int v = 0; v < 8; ++v) {
            const int r = v + (hi << 3);
            const size_t idx = ((size_t)(t0 + r) * B + b) * H2 + dir * H + col;
            ws[idx] = c[v];
        }
    }
}

// ---------------------------------------------------------------------------
// Kernel 2: serial scan. 16 WGs (2 dirs x 8 batch tiles) x 4 waves.
// h kept in LDS; W_hh fragments pinned in VGPRs; next step's xw prefetched.
// In-place: reads xw from ws, writes h back to the same slot.
// ---------------------------------------------------------------------------
__global__ void k_scan(const float* __restrict__ Whf,
                       const float* __restrict__ Whb,
                       float* __restrict__ ws) {
    const int lane  = threadIdx.x & 31;
    const int wv    = threadIdx.x >> 5;      // 0..3 -> n-tile
    const int dir   = blockIdx.x >> 3;
    const int b0    = (blockIdx.x & 7) * 16;
    const int mrow  = lane & 15;
    const int hi    = lane >> 4;
    const int khalf = hi << 1;
    const int col   = wv * 16 + mrow;        // h column this wave produces

    const float* W = dir ? Whb : Whf;

    // W_hh^T fragments: B[k][n] = W[col][k], K=64 -> 16 K-steps, pinned in VGPRs.
    v2f wf[16];
#pragma unroll
    for (int ks = 0; ks < 16; ++ks)
        wf[ks] = *(const v2f*)(W + (size_t)col * H + ks * 4 + khalf);

    constexpr int LST = 66;                  // padded row stride: conflict-free
    __shared__ float hbuf[16 * LST];
    for (int i = threadIdx.x; i < 16 * LST; i += blockDim.x) hbuf[i] = 0.f;
    __syncthreads();

    // Prefetch xw for the first step.
    float cnext[8];
    {
        const int tt0 = dir ? (T - 1) : 0;
        const float* p = ws + ((size_t)tt0 * B + b0) * H2 + dir * H;
#pragma unroll
        for (int v = 0; v < 8; ++v)
            cnext[v] = p[(size_t)(v + (hi << 3)) * H2 + col];
    }

    for (int step = 0; step < T; ++step) {
        const int tt = dir ? (T - 1 - step) : step;
        float* wsrow = ws + ((size_t)tt * B + b0) * H2 + dir * H;

        // Two parallel accumulation chains to shorten WMMA dependency latency.
        v8f ca, cb;
#pragma unroll
        for (int v = 0; v < 8; ++v) { ca[v] = cnext[v]; cb[v] = 0.f; }

        // Prefetch next step's xw (hits L2; overlaps the WMMA chain below).
        if (step + 1 < T) {
            const int ttn = dir ? (T - 2 - step) : (step + 1);
            const float* p = ws + ((size_t)ttn * B + b0) * H2 + dir * H;
#pragma unroll
            for (int v = 0; v < 8; ++v)
                cnext[v] = p[(size_t)(v + (hi << 3)) * H2 + col];
        }

        // A fragments: h_prev rows from LDS (transposed-read, padded stride).
        v2f af[16];
#pragma unroll
        for (int ks = 0; ks < 16; ++ks)
            af[ks] = *(const v2f*)(&hbuf[mrow * LST + ks * 4 + khalf]);

        __syncthreads();  // all waves done reading h_prev before overwrite

#pragma unroll
        for (int ks = 0; ks < 8; ++ks) {
            ca = wmma4(af[2 * ks],     wf[2 * ks],     ca);
            cb = wmma4(af[2 * ks + 1], wf[2 * ks + 1], cb);
        }

#pragma unroll
        for (int v = 0; v < 8; ++v) {
            const float h = fmaxf(ca[v] + cb[v], 0.f);   // relu
            const int r = v + (hi << 3);
            hbuf[r * LST + col] = h;                     // next step's input
            wsrow[(size_t)r * H2 + col] = h;             // persist for MLP head
        }
        __syncthreads();  // h_t visible to all waves
    }
}

// ---------------------------------------------------------------------------
// Kernel 3: MLP head. out[b*T+t] = lrelu(h[r,:] @ ff0^T + b0) . ff1 + b1
// One wave = 16 ws rows (same t, 16 batch rows). K=128, 8 N-tiles.
// ---------------------------------------------------------------------------
__global__ void k_head(const float* __restrict__ ws,
                       const float* __restrict__ ff0w, const float* __restrict__ ff0b,
                       const float* __restrict__ ff1w, const float* __restrict__ ff1b,
                       float* __restrict__ out) {
    const int lane  = threadIdx.x & 31;
    const int wid   = blockIdx.x * (blockDim.x >> 5) + (threadIdx.x >> 5); // 0..16383
    const int mrow  = lane & 15;
    const int hi    = lane >> 4;
    const int khalf = hi << 1;
    const int r0    = wid * 16;              // flat ws row base (= t*B + b)

    // A fragments: 16 rows x K=128 -> 32 K-steps, held in 64 VGPRs.
    v2f a[32];
    const float* hrow = ws + (size_t)(r0 + mrow) * H2;
#pragma unroll
    for (int ks = 0; ks < 32; ++ks)
        a[ks] = *(const v2f*)(hrow + ks * 4 + khalf);

    float acc[8];
#pragma unroll
    for (int v = 0; v < 8; ++v) acc[v] = 0.f;

#pragma unroll
    for (int nt = 0; nt < 8; ++nt) {
        const int col = nt * 16 + mrow;
        const float bias = ff0b[col];
        v8f c;
#pragma unroll
        for (int v = 0; v < 8; ++v) c[v] = bias;
#pragma unroll
        for (int ks = 0; ks < 32; ++ks) {
            v2f bw = *(const v2f*)(ff0w + (size_t)col * H2 + ks * 4 + khalf);
            c = wmma4(a[ks], bw, c);
        }
        const float w1 = ff1w[col];
#pragma unroll
        for (int v = 0; v < 8; ++v) {
            float y = c[v];
            y = (y > 0.f) ? y : 0.01f * y;   // leaky_relu
            acc[v] = fmaf(y, w1, acc[v]);
        }
    }

    // Reduce the 128 columns: butterfly within each 16-lane half.
#pragma unroll
    for (int off = 1; off < 16; off <<= 1)
#pragma unroll
        for (int v = 0; v < 8; ++v)
            acc[v] += __shfl_xor(acc[v], off, 32);

    if (mrow == 0) {
        const float b1 = ff1b[0];
#pragma unroll
        for (int v = 0; v < 8; ++v) {
            const int r = r0 + v + (hi << 3);      // flat row = t*B + b
            const int t = r >> 7;
            const int b = r & 127;
            out[(size_t)b * T + t] = acc[v] + b1;
        }
    }
}

// ---------------------------------------------------------------------------
extern "C" void kernel_launch(void* const* d_in, const int* in_sizes, int n_in,
                              void* d_out, int out_size, void* d_ws, size_t ws_size,
                              hipStream_t stream) {
    const float* x      = (const float*)d_in[0];
    const float* W_ih_f = (const float*)d_in[1];
    const float* W_hh_f = (const float*)d_in[2];
    const float* b_ih_f = (const float*)d_in[3];
    const float* b_hh_f = (const float*)d_in[4];
    const float* W_ih_b = (const float*)d_in[5];
    const float* W_hh_b = (const float*)d_in[6];
    const float* b_ih_b = (const float*)d_in[7];
    const float* b_hh_b = (const float*)d_in[8];
    const float* ff0w   = (const float*)d_in[9];
    const float* ff0b   = (const float*)d_in[10];
    const float* ff1w   = (const float*)d_in[11];
    const float* ff1b   = (const float*)d_in[12];

    float* ws  = (float*)d_ws;    // [T][B][H2] fp32 = 128 MB
    float* out = (float*)d_out;   // [B*T] fp32

    // Phase 1: xw for both directions (32768 waves, 8 waves/block).
    k_inproj<<<4096, 256, 0, stream>>>(x, W_ih_f, W_ih_b,
                                       b_ih_f, b_hh_f, b_ih_b, b_hh_b, ws);
    // Phase 2: serial bidirectional scan (16 WGs x 4 waves).
    k_scan<<<16, 128, 0, stream>>>(W_hh_f, W_hh_b, ws);
    // Phase 3: MLP head (16384 waves, 8 waves/block).
    k_head<<<2048, 256, 0, stream>>>(ws, ff0w, ff0b, ff1w, ff1b, out);
}